// LSSOT_72602127172320
// MI455X (gfx1250) — compile-verified
//
#include <hip/hip_runtime.h>
#include <math.h>

// ---------------- LSSOT pipeline for MI455X (gfx1250) ----------------
// Stage 1: row norms of x1                  (norm_kernel)
// Stage 2: thin QR of Z (Gram-Schmidt)      (qr_kernel)  -> packed V for WMMA B
// Stage 3: fp32 WMMA GEMM x1 @ V fused with theta/weight epilogue -> sort keys
// Stage 4: per-projection in-LDS bitonic sort + CDF scan + interp + cost
//          (keys staged into LDS with GLOBAL_LOAD_ASYNC_TO_LDS_B128)
// Stage 5: deterministic final reduction -> sqrt(mean)

typedef __attribute__((ext_vector_type(2))) float v2f;
typedef __attribute__((ext_vector_type(8))) float v8f;

#define NROWS   16384
#define NDIM    1024
#define NPROJ   512
#define NCOLS   1024          // 2 * NPROJ
#define NREF    1024
#define N3      3072          // 3 * NREF
#define PI_F    3.14159265358979323846f
#define INV2PI  0.15915494309189535f
#define EPS_F   1.1920929e-7f

#define KEYS_BYTES ((size_t)NPROJ * NROWS * 8)          // 64 MB
#define VB_BYTES   ((size_t)(NDIM/2) * NCOLS * 2 * 4)   // 4 MB
#define INVN_BYTES ((size_t)NROWS * 4)
#define PROJ_LDS   (NROWS * 8 + N3 * 4 + 1024 * 4)      // 147456 B < 160 KB

__device__ __forceinline__ float hi32f(unsigned long long k) {
    return __uint_as_float((unsigned)(k >> 32));
}
__device__ __forceinline__ float lo32f(unsigned long long k) {
    return __uint_as_float((unsigned)(k & 0xffffffffu));
}

// -------- Stage 1: inv_norm[n] = 1 / max(||x1[n]||, 1e-12) --------
__global__ __launch_bounds__(256) void lssot_norm_kernel(
    const float* __restrict__ X1, float* __restrict__ inv_norm)
{
    __shared__ float sred[256];
    const int n = blockIdx.x;
    float s = 0.0f;
    for (int d = threadIdx.x; d < NDIM; d += 256) {
        float v = X1[(size_t)n * NDIM + d];
        s += v * v;
    }
    sred[threadIdx.x] = s;
    __syncthreads();
    for (int off = 128; off > 0; off >>= 1) {
        if (threadIdx.x < off) sred[threadIdx.x] += sred[threadIdx.x + off];
        __syncthreads();
    }
    if (threadIdx.x == 0)
        inv_norm[n] = 1.0f / fmaxf(sqrtf(sred[0]), 1e-12f);
}

// -------- Stage 2: QR of Z (per-l 1024x2 Gram-Schmidt), write packed V --------
// Packed layout: VB[(p*NCOLS + c)*2 + e] = V[2p+e][c] where V[d][2l+k] = U[l][d][k]
__global__ __launch_bounds__(256) void lssot_qr_kernel(
    const float* __restrict__ Z, float* __restrict__ VB)
{
    __shared__ float r0s[256], r1s[256];
    const int l = blockIdx.x;
    const int t = threadIdx.x;
    float z0[4], z1[4];
    float s00 = 0.0f, s01 = 0.0f;
    for (int j = 0; j < 4; ++j) {
        int d = t * 4 + j;
        z0[j] = Z[(size_t)l * (NDIM * 2) + d * 2 + 0];
        z1[j] = Z[(size_t)l * (NDIM * 2) + d * 2 + 1];
        s00 += z0[j] * z0[j];
        s01 += z0[j] * z1[j];
    }
    r0s[t] = s00; r1s[t] = s01;
    __syncthreads();
    for (int off = 128; off > 0; off >>= 1) {
        if (t < off) { r0s[t] += r0s[t + off]; r1s[t] += r1s[t + off]; }
        __syncthreads();
    }
    const float S00 = r0s[0], S01 = r1s[0];
    __syncthreads();
    const float inv0 = 1.0f / sqrtf(fmaxf(S00, 1e-30f));
    const float proj = S01 / fmaxf(S00, 1e-30f);
    float v1[4];
    float s11 = 0.0f;
    for (int j = 0; j < 4; ++j) {
        v1[j] = z1[j] - proj * z0[j];
        s11 += v1[j] * v1[j];
    }
    r0s[t] = s11;
    __syncthreads();
    for (int off = 128; off > 0; off >>= 1) {
        if (t < off) r0s[t] += r0s[t + off];
        __syncthreads();
    }
    const float inv1 = 1.0f / sqrtf(fmaxf(r0s[0], 1e-30f));
    for (int j = 0; j < 4; ++j) {
        int d = t * 4 + j;
        int p = d >> 1;
        int e = d & 1;
        size_t base = ((size_t)p * NCOLS + 2 * l) * 2 + e;
        VB[base]     = z0[j] * inv0;   // column c = 2l   (u0)
        VB[base + 2] = v1[j] * inv1;   // column c = 2l+1 (u1)
    }
}

// -------- Stage 3: fp32 WMMA GEMM + theta/weight epilogue --------
// grid (128, 64), block 256 (8 waves). Wave w: rows (bx*8+w)*16, cols by*16.
__global__ __launch_bounds__(256) void lssot_gemm_theta_kernel(
    const float* __restrict__ X1, const float* __restrict__ VB,
    const float* __restrict__ inv_norm, const float* __restrict__ W,
    unsigned long long* __restrict__ keys)
{
    const int wave = threadIdx.x >> 5;
    const int lane = threadIdx.x & 31;
    const int half = lane >> 4;
    const int lr   = lane & 15;
    const int row0 = (blockIdx.x * 8 + wave) * 16;
    const int col0 = blockIdx.y * 16;

    // A layout (16x4 f32): lanes 0-15 hold K=0,1 (v0,v1); lanes 16-31 K=2,3.
    // B packed so each lane loads (V[kk+2h][c], V[kk+2h+1][c]) as one float2.
    const float* arow = X1 + (size_t)(row0 + lr) * NDIM + 2 * half;
    const float* bcol = VB + (size_t)half * (NCOLS * 2) + (size_t)(col0 + lr) * 2;

    v8f acc = {};
#pragma unroll 4
    for (int kk = 0; kk < NDIM; kk += 4) {
        v2f a = *(const v2f*)(arow + kk);
        v2f b = *(const v2f*)(bcol + (size_t)(kk >> 1) * (NCOLS * 2));
        acc = __builtin_amdgcn_wmma_f32_16x16x4_f32(
            false, a, false, b, (short)0, acc, false, false);
    }

    // Epilogue: adjacent columns (2l, 2l+1) sit in adjacent lanes.
    // Even lane owns p0, partner (xor 1) supplies p1.
    const int l = (col0 + (lr & ~1)) >> 1;
#pragma unroll
    for (int r = 0; r < 8; ++r) {
        float own   = acc[r];
        float other = __shfl_xor(own, 1, 32);
        if ((lr & 1) == 0) {
            const int n   = row0 + half * 8 + r;
            const float p0 = own, p1 = other;
            const float invn = inv_norm[n];
            const float nrm  = sqrtf(p0 * p0 + p1 * p1) * invn;
            float theta = (atan2f(-p1, -p0) + PI_F) * INV2PI;
            float mw = (nrm <= 1e-6f) ? 0.0f : W[n];
            unsigned long long key =
                ((unsigned long long)__float_as_uint(theta) << 32) |
                (unsigned long long)__float_as_uint(mw);
            keys[(size_t)l * NROWS + n] = key;
        }
    }
}

// -------- Stage 4: per-projection sort + CDF + interp + cost --------
__global__ __launch_bounds__(1024) void lssot_proj_kernel(
    const unsigned long long* __restrict__ keys, float* __restrict__ sums)
{
    extern __shared__ unsigned char smem[];
    unsigned long long* K = (unsigned long long*)smem;            // 16384 keys
    float* F   = (float*)(smem + NROWS * 8);                      // 3072 F_vals
    float* red = (float*)(smem + NROWS * 8 + N3 * 4);             // 1024 scratch

    const int l   = blockIdx.x;
    const int tid = threadIdx.x;
    const int T   = 1024;

    // Stage keys into LDS via CDNA5 async-to-LDS DMA path (no VGPR round trip).
    // Each lane moves 16 B (two keys); tracked by ASYNCcnt.
    for (int i = tid * 2; i < NROWS; i += T * 2) {
        unsigned ldsoff = (unsigned)(unsigned long long)(&K[i]);
        unsigned long long gaddr =
            (unsigned long long)(keys + (size_t)l * NROWS + i);
        asm volatile("global_load_async_to_lds_b128 %0, %1, off"
                     :: "v"(ldsoff), "v"(gaddr) : "memory");
    }
    asm volatile("s_wait_asynccnt 0x0" ::: "memory");
    __syncthreads();

    // Bitonic sort ascending on packed (theta | weight) uint64
    for (int k = 2; k <= NROWS; k <<= 1) {
        for (int j = k >> 1; j > 0; j >>= 1) {
            for (int p = tid; p < NROWS / 2; p += T) {
                int i = ((p & ~(j - 1)) << 1) | (p & (j - 1));
                int m = i | j;
                bool up = ((i & k) == 0);
                unsigned long long a = K[i], b = K[m];
                if ((a > b) == up) { K[i] = b; K[m] = a; }
            }
            __syncthreads();
        }
    }

    // alpha = sum(theta*mw)/sum(mw) - 0.5
    float s1 = 0.0f, s0 = 0.0f;
    for (int i = tid; i < NROWS; i += T) {
        unsigned long long kk = K[i];
        float th = hi32f(kk), mw = lo32f(kk);
        s1 += th * mw;
        s0 += mw;
    }
    red[tid] = s1; __syncthreads();
    for (int off = 512; off > 0; off >>= 1) {
        if (tid < off) red[tid] += red[tid + off];
        __syncthreads();
    }
    const float S1 = red[0]; __syncthreads();
    red[tid] = s0; __syncthreads();
    for (int off = 512; off > 0; off >>= 1) {
        if (tid < off) red[tid] += red[tid + off];
        __syncthreads();
    }
    const float S0 = red[0]; __syncthreads();
    const float alpha = S1 / S0 - 0.5f;

    // Inclusive scan of sorted weights -> CDF, written back into low words
    const int base = tid * 16;
    float loc[16];
    float run = 0.0f;
    for (int j = 0; j < 16; ++j) {
        run += lo32f(K[base + j]);
        loc[j] = run;
    }
    red[tid] = run; __syncthreads();
    for (int off = 1; off < 1024; off <<= 1) {
        float v = red[tid];
        float add = (tid >= off) ? red[tid - off] : 0.0f;
        __syncthreads();
        red[tid] = v + add;
        __syncthreads();
    }
    const float exc = (tid == 0) ? 0.0f : red[tid - 1];
    __syncthreads();
    for (int j = 0; j < 16; ++j) {
        float cdf = loc[j] + exc;
        K[base + j] = (K[base + j] & 0xFFFFFFFF00000000ull) |
                      (unsigned long long)__float_as_uint(cdf);
    }
    __syncthreads();

    // F_vals over periodic extension: F(x+k) = k + interp(xs, cdf, frac)
    for (int jj = tid; jj < N3; jj += T) {
        float xq = -1.0f + 3.0f * (float)jj / 3071.0f;
        float fi = floorf(xq);
        float rest = xq - fi;
        int lo = 0, hi = NROWS;
        while (lo < hi) {                      // searchsorted left on theta
            int mid = (lo + hi) >> 1;
            if (hi32f(K[mid]) < rest) lo = mid + 1; else hi = mid;
        }
        int ind = min(max(lo - 1, 0), NROWS - 2);
        float x0 = hi32f(K[ind]), x1v = hi32f(K[ind + 1]);
        float y0 = lo32f(K[ind]), y1  = lo32f(K[ind + 1]);
        F[jj] = fi + y0 + (y1 - y0) / (x1v - x0 + EPS_F) * (rest - x0);
    }
    __syncthreads();

    // Inverse CDF at ref - alpha, accumulate LCOT cost
    float contrib = 0.0f;
    for (int ii = tid; ii < NREF; ii += T) {
        float x = (float)ii / (float)NREF;
        float q = x - alpha;
        int lo = 0, hi = N3;
        while (lo < hi) {                      // searchsorted left on F_vals
            int mid = (lo + hi) >> 1;
            if (F[mid] < q) lo = mid + 1; else hi = mid;
        }
        int ind = min(max(lo - 1, 0), N3 - 2);
        float F0 = F[ind], F1 = F[ind + 1];
        float xn0 = -1.0f + 3.0f * (float)ind / 3071.0f;
        float xn1 = -1.0f + 3.0f * (float)(ind + 1) / 3071.0f;
        float val = xn0 + (xn1 - xn0) / (F1 - F0 + EPS_F) * (q - F0);
        float e = val - x;
        float a = fabsf(e);
        float m = fminf(a, 1.0f - a);
        contrib += m * m;
    }
    red[tid] = contrib; __syncthreads();
    for (int off = 512; off > 0; off >>= 1) {
        if (tid < off) red[tid] += red[tid + off];
        __syncthreads();
    }
    if (tid == 0) sums[l] = red[0];
}

// -------- Stage 5: deterministic final reduction --------
__global__ __launch_bounds__(256) void lssot_final_kernel(
    const float* __restrict__ sums, float* __restrict__ out)
{
    __shared__ float sred[256];
    float s = 0.0f;
    for (int i = threadIdx.x; i < NPROJ; i += 256) s += sums[i];
    sred[threadIdx.x] = s;
    __syncthreads();
    for (int off = 128; off > 0; off >>= 1) {
        if (threadIdx.x < off) sred[threadIdx.x] += sred[threadIdx.x + off];
        __syncthreads();
    }
    if (threadIdx.x == 0) out[0] = sqrtf(sred[0] / (float)NPROJ);
}

extern "C" void kernel_launch(void* const* d_in, const int* in_sizes, int n_in,
                              void* d_out, int out_size, void* d_ws, size_t ws_size,
                              hipStream_t stream) {
    const float* x1 = (const float*)d_in[0];   // (16384, 1024)
    const float* w  = (const float*)d_in[1];   // (16384,)
    const float* Z  = (const float*)d_in[2];   // (512, 1024, 2)
    float* out = (float*)d_out;

    char* ws = (char*)d_ws;
    unsigned long long* keys = (unsigned long long*)ws;                 // 64 MB
    float* VB   = (float*)(ws + KEYS_BYTES);                            // 4 MB
    float* invn = (float*)(ws + KEYS_BYTES + VB_BYTES);                 // 64 KB
    float* sums = (float*)(ws + KEYS_BYTES + VB_BYTES + INVN_BYTES);    // 2 KB

    lssot_norm_kernel<<<NROWS, 256, 0, stream>>>(x1, invn);
    lssot_qr_kernel<<<NPROJ, 256, 0, stream>>>(Z, VB);
    lssot_gemm_theta_kernel<<<dim3(NROWS / 128, NCOLS / 16), 256, 0, stream>>>(
        x1, VB, invn, w, keys);
    lssot_proj_kernel<<<NPROJ, 1024, PROJ_LDS, stream>>>(keys, sums);
    lssot_final_kernel<<<1, 256, 0, stream>>>(sums, out);
}